// MultiClassGHMCLoss_11123965296941
// MI455X (gfx1250) — compile-verified
//
#include <hip/hip_runtime.h>
#include <hip/hip_bf16.h>
#include <math.h>

#define BINS 30
#define WAVES_PER_BLOCK 8
#define NBUF4 252   // padded float4 slots per row buffer (1008 floats >= C=1000)

typedef int v4i __attribute__((ext_vector_type(4)));
typedef __attribute__((address_space(1))) v4i* gptr128_t;  // global AS
typedef __attribute__((address_space(3))) v4i* lptr128_t;  // LDS AS

// ---- CDNA5 async global->LDS copy (16B per lane per issue) ----------------
__device__ __forceinline__ void async_copy_b128(const float4* g, float4* l) {
#if __has_builtin(__builtin_amdgcn_global_load_async_to_lds_b128)
  __builtin_amdgcn_global_load_async_to_lds_b128(
      (gptr128_t)(float4*)g,   // drop const, addrspacecast to AS1, bitcast to v4i
      (lptr128_t)l,            // addrspacecast to AS3, bitcast to v4i
      /*offset=*/0, /*cpol=*/0);
#else
  unsigned lds_addr =
      (unsigned)(unsigned long long)(__attribute__((address_space(3))) void*)l;
  asm volatile("global_load_async_to_lds_b128 %0, %1, off"
               :: "v"(lds_addr), "v"(g) : "memory");
#endif
}

__device__ __forceinline__ void wait_async_le8() {
#if __has_builtin(__builtin_amdgcn_s_wait_asynccnt)
  __builtin_amdgcn_s_wait_asynccnt(8);
#else
  asm volatile("s_wait_asynccnt 8" ::: "memory");
#endif
}

__device__ __forceinline__ void wait_async_0() {
#if __has_builtin(__builtin_amdgcn_s_wait_asynccnt)
  __builtin_amdgcn_s_wait_asynccnt(0);
#else
  asm volatile("s_wait_asynccnt 0" ::: "memory");
#endif
}

// ---- main streaming kernel: one row per wave, async double-buffered -------
__global__ __launch_bounds__(256) void ghm_rows_kernel(
    const float* __restrict__ preds, const int* __restrict__ targets,
    int C, int B, float* __restrict__ ws) {
  __shared__ float4 sbuf[2 * WAVES_PER_BLOCK * NBUF4];  // 64512 bytes
  __shared__ float h_cnt[BINS];
  __shared__ float h_nll[BINS];

  const int tid  = threadIdx.x;
  const int lane = tid & 31;
  const int widx = tid >> 5;
  if (tid < BINS) { h_cnt[tid] = 0.f; h_nll[tid] = 0.f; }
  __syncthreads();

  const int nf4        = C >> 2;              // 250 for C=1000
  const int padded     = (nf4 + 31) & ~31;    // 256
  const int stage_ops  = padded >> 5;         // 8
  const int total_waves = gridDim.x * WAVES_PER_BLOCK;
  const int wave_id     = blockIdx.x * WAVES_PER_BLOCK + widx;

  float4* const buf0 = &sbuf[widx * NBUF4];
  float4* const buf1 = &sbuf[(WAVES_PER_BLOCK + widx) * NBUF4];

  int cur = 0;
  int row = wave_id;
  if (row < B) {
    const float4* g = (const float4*)preds + (size_t)row * nf4;
    for (int i = lane; i < padded; i += 32) {
      int idx = i < nf4 ? i : (nf4 - 1);      // tail lanes duplicate last slot
      async_copy_b128(g + idx, buf0 + idx);
    }
  }

  for (; row < B; row += total_waves) {
    const int nrow = row + total_waves;
    float4* const lcur = cur ? buf1 : buf0;
    float4* const lnxt = cur ? buf0 : buf1;

    if (nrow < B) {
      const float4* g = (const float4*)preds + (size_t)nrow * nf4;
      for (int i = lane; i < padded; i += 32) {
        int idx = i < nf4 ? i : (nf4 - 1);
        async_copy_b128(g + idx, lnxt + idx);
      }
      if (stage_ops == 8) wait_async_le8();   // current row done, prefetch in flight
      else                wait_async_0();     // generic fallback
    } else {
      wait_async_0();
    }

    // pass 1: row max (from LDS)
    float m = -INFINITY;
    for (int j = lane; j < nf4; j += 32) {
      float4 v = lcur[j];
      m = fmaxf(m, fmaxf(fmaxf(v.x, v.y), fmaxf(v.z, v.w)));
    }
#pragma unroll
    for (int off = 16; off; off >>= 1)
      m = fmaxf(m, __shfl_xor(m, off, 32));

    // pass 2: sum of exp(x - m) (from LDS)
    float s = 0.f;
    for (int j = lane; j < nf4; j += 32) {
      float4 v = lcur[j];
      s += __expf(v.x - m) + __expf(v.y - m) + __expf(v.z - m) + __expf(v.w - m);
    }
#pragma unroll
    for (int off = 16; off; off >>= 1)
      s += __shfl_xor(s, off, 32);

    if (lane == 0) {
      const int   t  = targets[row];
      const float xt = ((const float*)lcur)[t];
      float p = __expf(xt - m) / s;
      p = fminf(fmaxf(p, 1e-10f), 1.f - 1e-10f);
      const float gval = 1.f - p;                     // |p_true - 1|
      int bin = (int)(gval * (float)BINS);
      bin = bin < 0 ? 0 : (bin > BINS - 1 ? BINS - 1 : bin);
      atomicAdd(&h_cnt[bin], 1.f);                    // ds_add_f32
      atomicAdd(&h_nll[bin], -__logf(p));
    }
    cur ^= 1;
  }

  __syncthreads();
  if (tid < BINS) {
    const float c = h_cnt[tid];
    if (c > 0.f) {                                    // only nonempty bins hit HBM
      atomicAdd(&ws[tid], c);
      atomicAdd(&ws[BINS + tid], h_nll[tid]);
    }
  }
}

// ---- scratch init (must run every call; harness does not re-zero ws) ------
__global__ void ghm_zero_kernel(float* __restrict__ ws) {
  if (threadIdx.x < 2 * BINS) ws[threadIdx.x] = 0.f;
}

// ---- finalize: EMA update + weighting + scalar loss (one wave) ------------
__global__ void ghm_finalize_kernel(const float* __restrict__ ws,
                                    const float* __restrict__ acc_sum,
                                    int B, float* __restrict__ out) {
  const int i = threadIdx.x;
  float cnt = 0.f, nll = 0.f, acc = 0.f;
  if (i < BINS) { cnt = ws[i]; nll = ws[BINS + i]; acc = acc_sum[i]; }
  const bool  ne   = (i < BINS) && (cnt > 0.f);
  const float nacc = ne ? (0.75f * acc + 0.25f * cnt) : acc;
  const float w    = ne ? ((float)B / fmaxf(nacc, 1e-10f)) : 0.f;
  float contrib = w * nll;
  float nfl     = ne ? 1.f : 0.f;
#pragma unroll
  for (int off = 16; off; off >>= 1) {
    contrib += __shfl_xor(contrib, off, 32);
    nfl     += __shfl_xor(nfl, off, 32);
  }
  if (i == 0)
    out[0] = (contrib / fmaxf(nfl, 1.f)) / (float)B * 1.0f;  // LOSS_WEIGHT = 1
}

extern "C" void kernel_launch(void* const* d_in, const int* in_sizes, int n_in,
                              void* d_out, int out_size, void* d_ws, size_t ws_size,
                              hipStream_t stream) {
  const float* preds   = (const float*)d_in[0];
  const int*   targets = (const int*)d_in[1];
  const float* acc_sum = (const float*)d_in[2];
  float*       out     = (float*)d_out;
  float*       ws      = (float*)d_ws;

  const int B = in_sizes[1];             // 131072
  const int C = in_sizes[0] / B;         // 1000

  ghm_zero_kernel<<<1, 64, 0, stream>>>(ws);

  int blocks = (B + WAVES_PER_BLOCK - 1) / WAVES_PER_BLOCK;
  if (blocks > 2048) blocks = 2048;      // 16384 persistent waves, 8 rows each
  ghm_rows_kernel<<<blocks, 256, 0, stream>>>(preds, targets, C, B, ws);

  ghm_finalize_kernel<<<1, 32, 0, stream>>>(ws, acc_sum, B, out);
}